// BetaModel_8040178778113
// MI455X (gfx1250) — compile-verified
//
#include <hip/hip_runtime.h>
#include <hip/hip_bf16.h>
#include <math.h>

// ---------------------------------------------------------------------------
// Mamba-2-layer + MLP head, fp32 end-to-end.
// GEMMs run on the CDNA5 matrix pipe via V_WMMA_F32_16X16X4_F32 (fp32 WMMA,
// keeps reference numerics; workload is bandwidth/latency bound so low-precision
// WMMA buys nothing). Each wave register-blocks 2 adjacent N tiles (reusing its
// A fragment) when N % 32 == 0. Selective scan: 1024 wave-parallel recurrences,
// lane = state index.
// ---------------------------------------------------------------------------

#define BLT   4096   // B*L tokens
#define SEQ   2048
#define DM    256
#define DI    512
#define DST   32
#define DTR   16

typedef __attribute__((ext_vector_type(2))) float v2f;
typedef __attribute__((ext_vector_type(8))) float v8f;

// ---------------------------------------------------------------------------
// Generic WMMA GEMM:  C[M,N] = A[M,K] * W[N,K]^T  (+ bias[n], activation)
// act: 0 = none, 1 = relu(x+bias), 2 = softplus(x+bias)
// One wave computes a 16 x (16*NB) tile; K consumed 4/step by wmma_f32_16x16x4.
// A-frag (16x4 f32): lane L -> row m0+(L&15), K pair (L>>4)*2   (8B load)
// B-frag (4x16 f32): lane L -> col n0+(L&15), K pair (L>>4)*2   (8B load, W rows)
// C/D   : VGPR i -> row m0 + i + (L>>4)*8, col n0+(L&15)
// ---------------------------------------------------------------------------
template <int NB>
__global__ __launch_bounds__(128)
void wmma_gemm_nt(const float* __restrict__ A, int lda,
                  const float* __restrict__ W, int ldw,
                  float* __restrict__ C, int ldc,
                  int M, int N, int K,
                  const float* __restrict__ bias, int act)
{
    const int lane = threadIdx.x;      // 0..31
    const int r    = lane & 15;
    const int kh   = lane >> 4;        // 0 or 1
    const int m0   = blockIdx.x * 16;
    const int n0   = (blockIdx.y * blockDim.y + threadIdx.y) * (16 * NB);
    if (m0 >= M || n0 >= N) return;    // whole-wave uniform guard (EXEC stays all-1)

    const float* Ap = A + (size_t)(m0 + r) * lda + kh * 2;
    const float* Wp[NB];
    #pragma unroll
    for (int j = 0; j < NB; ++j)
        Wp[j] = W + (size_t)(n0 + j * 16 + r) * ldw + kh * 2;

    v8f acc[NB];
    #pragma unroll
    for (int j = 0; j < NB; ++j) acc[j] = (v8f){0.f,0.f,0.f,0.f,0.f,0.f,0.f,0.f};

    #pragma unroll 4
    for (int k0 = 0; k0 < K; k0 += 4) {
        v2f a = *(const v2f*)(Ap + k0);
        #pragma unroll
        for (int j = 0; j < NB; ++j) {
            v2f b = *(const v2f*)(Wp[j] + k0);
            acc[j] = __builtin_amdgcn_wmma_f32_16x16x4_f32(
                /*neg_a=*/false, a, /*neg_b=*/false, b,
                /*c_mod=*/(short)0, acc[j], /*reuse_a=*/false, /*reuse_b=*/false);
        }
    }

    #pragma unroll
    for (int j = 0; j < NB; ++j) {
        const int col = n0 + j * 16 + r;
        const float bv = bias ? bias[col] : 0.0f;
        #pragma unroll
        for (int i = 0; i < 8; ++i) {
            float v = acc[j][i] + bv;
            if (act == 1)      v = fmaxf(v, 0.0f);
            else if (act == 2) v = fmaxf(v, 0.0f) + log1pf(expf(-fabsf(v))); // softplus
            C[(size_t)(m0 + i + kh * 8) * ldc + col] = v;
        }
    }
}

// --------------------------- input max (x[...,1], x[...,2]) -----------------
__global__ void input_max_kernel(const float* __restrict__ x, float* __restrict__ scal)
{
    __shared__ float s1[256], s2[256];
    int tid = threadIdx.x;
    float m1 = -3.4e38f, m2 = -3.4e38f;
    for (int t = tid; t < BLT; t += 256) {
        m1 = fmaxf(m1, x[t * 3 + 1]);
        m2 = fmaxf(m2, x[t * 3 + 2]);
    }
    s1[tid] = m1; s2[tid] = m2; __syncthreads();
    for (int s = 128; s > 0; s >>= 1) {
        if (tid < s) { s1[tid] = fmaxf(s1[tid], s1[tid + s]); s2[tid] = fmaxf(s2[tid], s2[tid + s]); }
        __syncthreads();
    }
    if (tid == 0) { scal[0] = s1[0]; scal[1] = s2[0]; }
}

// --------------------- normalize + fc (3 -> 256) ---------------------------
__global__ void pre_fc_kernel(const float* __restrict__ x,
                              const float* __restrict__ fcw,
                              const float* __restrict__ fcb,
                              const float* __restrict__ scal,
                              float* __restrict__ h)
{
    int idx = blockIdx.x * blockDim.x + threadIdx.x;  // BLT*DM
    if (idx >= BLT * DM) return;
    int t = idx >> 8, d = idx & 255;
    float x0 = x[t * 3 + 0] * (1.0f / 255.0f);
    float x1 = x[t * 3 + 1] / (scal[0] + 1e-8f);
    float x2 = x[t * 3 + 2] / (scal[1] + 1e-8f);
    h[idx] = x0 * fcw[d * 3] + x1 * fcw[d * 3 + 1] + x2 * fcw[d * 3 + 2] + fcb[d];
}

// -------------- depthwise causal conv(4) + bias + SiLU ---------------------
__global__ void conv_silu_kernel(const float* __restrict__ xz,
                                 const float* __restrict__ cw,
                                 const float* __restrict__ cb,
                                 float* __restrict__ xin)
{
    int idx = blockIdx.x * blockDim.x + threadIdx.x;  // BLT*DI
    if (idx >= BLT * DI) return;
    int t = idx >> 9, d = idx & 511;
    int l = t & (SEQ - 1);
    float acc = cb[d];
    #pragma unroll
    for (int j = 0; j < 4; ++j) {
        int li = l - 3 + j;
        if (li >= 0)
            acc = fmaf(xz[(size_t)(t - 3 + j) * (2 * DI) + d], cw[d * 4 + j], acc);
    }
    xin[idx] = acc / (1.0f + expf(-acc));   // SiLU
}

// ----------------- selective scan + D-skip + SiLU(z) gate -------------------
// One wave per (b,d) chain; lane = state index n (0..31). Sequential over L.
__global__ void scan_kernel(const float* __restrict__ delta,
                            const float* __restrict__ xin,
                            const float* __restrict__ dbc,
                            const float* __restrict__ xz,
                            const float* __restrict__ Alog,
                            const float* __restrict__ Dp,
                            float* __restrict__ y)
{
    int wave = blockIdx.x * blockDim.y + threadIdx.y;   // 0..1023
    int lane = threadIdx.x;                             // state n
    int b = wave >> 9;
    int d = wave & 511;
    float Aval = -expf(Alog[d * DST + lane]);
    float Dval = Dp[d];
    float state = 0.0f;
    for (int l = 0; l < SEQ; ++l) {
        int t = b * SEQ + l;
        float dlt = delta[(size_t)t * DI + d];
        float xv  = xin[(size_t)t * DI + d];
        float Bv  = dbc[(size_t)t * 80 + DTR + lane];        // B row, coalesced
        float Cv  = dbc[(size_t)t * 80 + DTR + DST + lane];  // C row, coalesced
        float da  = expf(dlt * Aval);
        state = fmaf(da, state, dlt * xv * Bv);
        float c = state * Cv;
        #pragma unroll
        for (int off = 16; off > 0; off >>= 1)
            c += __shfl_xor(c, off, 32);
        if (lane == 0) {
            float zv = xz[(size_t)t * (2 * DI) + DI + d];
            float g  = zv / (1.0f + expf(-zv));              // SiLU gate
            y[(size_t)t * DI + d] = (c + Dval * xv) * g;
        }
    }
}

// ------------------------- global min/max (2 stages) ------------------------
__global__ void minmax_partial(const float* __restrict__ v, int n,
                               float* __restrict__ pmin, float* __restrict__ pmax)
{
    __shared__ float smin[256], smax[256];
    int tid = threadIdx.x;
    float lo = 3.4e38f, hi = -3.4e38f;
    int base = blockIdx.x * 4096;
    for (int j = 0; j < 16; ++j) {
        int idx = base + j * 256 + tid;
        if (idx < n) { float x = v[idx]; lo = fminf(lo, x); hi = fmaxf(hi, x); }
    }
    smin[tid] = lo; smax[tid] = hi; __syncthreads();
    for (int s = 128; s > 0; s >>= 1) {
        if (tid < s) { smin[tid] = fminf(smin[tid], smin[tid + s]);
                       smax[tid] = fmaxf(smax[tid], smax[tid + s]); }
        __syncthreads();
    }
    if (tid == 0) { pmin[blockIdx.x] = smin[0]; pmax[blockIdx.x] = smax[0]; }
}

__global__ void minmax_final(const float* __restrict__ pmin,
                             const float* __restrict__ pmax,
                             int nb, float* __restrict__ scal)
{
    __shared__ float smin[256], smax[256];
    int tid = threadIdx.x;
    float lo = 3.4e38f, hi = -3.4e38f;
    for (int i = tid; i < nb; i += 256) { lo = fminf(lo, pmin[i]); hi = fmaxf(hi, pmax[i]); }
    smin[tid] = lo; smax[tid] = hi; __syncthreads();
    for (int s = 128; s > 0; s >>= 1) {
        if (tid < s) { smin[tid] = fminf(smin[tid], smin[tid + s]);
                       smax[tid] = fmaxf(smax[tid], smax[tid + s]); }
        __syncthreads();
    }
    if (tid == 0) { scal[0] = smin[0]; scal[1] = smax[0]; }
}

// ----------------- [-1,1] rescale + per-token softmax(256) ------------------
__global__ void softmax_kernel(const float* __restrict__ hin,
                               const float* __restrict__ scal,
                               float* __restrict__ out)
{
    __shared__ float sm[256];
    int t = blockIdx.x, i = threadIdx.x;
    float gmin = scal[0], gmax = scal[1];
    float range = gmax - gmin;
    float v = hin[(size_t)t * DM + i];
    float hn = (range == 0.0f) ? 0.0f : (2.0f * (v - gmin) / range - 1.0f);
    sm[i] = hn; __syncthreads();
    for (int s = 128; s > 0; s >>= 1) {
        if (i < s) sm[i] = fmaxf(sm[i], sm[i + s]);
        __syncthreads();
    }
    float rmax = sm[0]; __syncthreads();
    float e = expf(hn - rmax);
    sm[i] = e; __syncthreads();
    for (int s = 128; s > 0; s >>= 1) {
        if (i < s) sm[i] += sm[i + s];
        __syncthreads();
    }
    out[(size_t)t * DM + i] = e / (sm[0] + 1e-8f);
}

// ---------------------------------------------------------------------------
static inline void launch_gemm(const float* A, int lda, const float* W, int ldw,
                               float* C, int ldc, int M, int N, int K,
                               const float* bias, int act, hipStream_t stream)
{
    dim3 block(32, 4);
    if (N % 32 == 0) {
        dim3 grid(M / 16, (N / 32 + 3) / 4);
        hipLaunchKernelGGL(wmma_gemm_nt<2>, grid, block, 0, stream,
                           A, lda, W, ldw, C, ldc, M, N, K, bias, act);
    } else {
        dim3 grid(M / 16, ((N + 15) / 16 + 3) / 4);
        hipLaunchKernelGGL(wmma_gemm_nt<1>, grid, block, 0, stream,
                           A, lda, W, ldw, C, ldc, M, N, K, bias, act);
    }
}

extern "C" void kernel_launch(void* const* d_in, const int* in_sizes, int n_in,
                              void* d_out, int out_size, void* d_ws, size_t ws_size,
                              hipStream_t stream)
{
    const float* x         = (const float*)d_in[0];
    const float* fc_w      = (const float*)d_in[1];
    const float* fc_b      = (const float*)d_in[2];
    const float* in_proj_w = (const float*)d_in[3];   // (2, 1024, 256)
    const float* conv_w    = (const float*)d_in[4];   // (2, 512, 4)
    const float* conv_b    = (const float*)d_in[5];   // (2, 512)
    const float* x_proj_w  = (const float*)d_in[6];   // (2, 80, 512)
    const float* dt_proj_w = (const float*)d_in[7];   // (2, 512, 16)
    const float* dt_proj_b = (const float*)d_in[8];   // (2, 512)
    const float* A_log     = (const float*)d_in[9];   // (2, 512, 32)
    const float* D_skip    = (const float*)d_in[10];  // (2, 512)
    const float* out_proj_w= (const float*)d_in[11];  // (2, 256, 512)
    const float* w1 = (const float*)d_in[12]; const float* b1 = (const float*)d_in[13];
    const float* w2 = (const float*)d_in[14]; const float* b2 = (const float*)d_in[15];
    const float* w3 = (const float*)d_in[16]; const float* b3 = (const float*)d_in[17];
    const float* w4 = (const float*)d_in[18]; const float* b4 = (const float*)d_in[19];
    float* out = (float*)d_out;

    // workspace layout (floats); total ~11.87M floats (~47.5 MB)
    float* ws    = (float*)d_ws;
    float* h     = ws;                    // BLT*DM      = 1,048,576
    float* xz    = h     + 1048576;       // BLT*2*DI    = 4,194,304
    float* xin   = xz    + 4194304;       // BLT*DI      = 2,097,152
    float* dbc   = xin   + 2097152;       // BLT*80      =   327,680
    float* delta = dbc   + 327680;        // BLT*DI      = 2,097,152
    float* yb    = delta + 2097152;       // BLT*DI      = 2,097,152
    float* red   = yb    + 2097152;       // 256+256+4
    float* pmin  = red;
    float* pmax  = red + 256;
    float* scal_in = red + 512;           // input maxima [2]
    float* scal_mm = red + 514;           // global min/max [2]
    // MLP temporaries alias the (dead-after-scan) xz region
    float* m1 = xz;            // BLT*64
    float* m2 = xz + 262144;   // BLT*64
    float* m3 = xz + 524288;   // BLT*64
    float* m4 = xz + 786432;   // BLT*DM

    // 1) input channel maxima, then normalize + fc -> h
    hipLaunchKernelGGL(input_max_kernel, dim3(1), dim3(256), 0, stream, x, scal_in);
    hipLaunchKernelGGL(pre_fc_kernel, dim3(BLT * DM / 256), dim3(256), 0, stream,
                       x, fc_w, fc_b, scal_in, h);

    // 2) two Mamba layers
    for (int i = 0; i < 2; ++i) {
        const float* Wi  = in_proj_w  + (size_t)i * (2 * DI) * DM;
        const float* cw  = conv_w     + (size_t)i * DI * 4;
        const float* cb  = conv_b     + (size_t)i * DI;
        const float* Wx  = x_proj_w   + (size_t)i * 80 * DI;
        const float* Wdt = dt_proj_w  + (size_t)i * DI * DTR;
        const float* bdt = dt_proj_b  + (size_t)i * DI;
        const float* Al  = A_log      + (size_t)i * DI * DST;
        const float* Dp  = D_skip     + (size_t)i * DI;
        const float* Wo  = out_proj_w + (size_t)i * DM * DI;

        // xz = h @ Wi^T                      (4096 x 1024 x 256)
        launch_gemm(h, DM, Wi, DM, xz, 2 * DI, BLT, 2 * DI, DM, nullptr, 0, stream);
        // xin = silu(depthwise_conv(xz[:, :512]) + cb)
        hipLaunchKernelGGL(conv_silu_kernel, dim3(BLT * DI / 256), dim3(256), 0, stream,
                           xz, cw, cb, xin);
        // dbc = xin @ Wx^T                   (4096 x 80 x 512)
        launch_gemm(xin, DI, Wx, DI, dbc, 80, BLT, 80, DI, nullptr, 0, stream);
        // delta = softplus(dbc[:, :16] @ Wdt^T + bdt)   (4096 x 512 x 16)
        launch_gemm(dbc, 80, Wdt, DTR, delta, DI, BLT, DI, DTR, bdt, 2, stream);
        // selective scan + gate -> yb
        hipLaunchKernelGGL(scan_kernel, dim3(128), dim3(32, 8), 0, stream,
                           delta, xin, dbc, xz, Al, Dp, yb);
        // h = yb @ Wo^T                      (4096 x 256 x 512)
        launch_gemm(yb, DI, Wo, DI, h, DM, BLT, DM, DI, nullptr, 0, stream);
    }

    // 3) MLP head (all relu(x+b))
    launch_gemm(h,  DM, w1, DM, m1, 64, BLT, 64, DM, b1, 1, stream);
    launch_gemm(m1, 64, w2, 64, m2, 64, BLT, 64, 64, b2, 1, stream);
    launch_gemm(m2, 64, w3, 64, m3, 64, BLT, 64, 64, b3, 1, stream);
    launch_gemm(m3, 64, w4, 64, m4, DM, BLT, DM, 64, b4, 1, stream);

    // 4) global min/max, rescale to [-1,1], softmax over last dim
    hipLaunchKernelGGL(minmax_partial, dim3(256), dim3(256), 0, stream,
                       m4, BLT * DM, pmin, pmax);
    hipLaunchKernelGGL(minmax_final, dim3(1), dim3(256), 0, stream,
                       pmin, pmax, 256, scal_mm);
    hipLaunchKernelGGL(softmax_kernel, dim3(BLT), dim3(256), 0, stream,
                       m4, scal_mm, out);
}